// MultiHeadAttention_9345848836102
// MI455X (gfx1250) — compile-verified
//
#include <hip/hip_runtime.h>

typedef float v2f __attribute__((ext_vector_type(2)));
typedef float v8f __attribute__((ext_vector_type(8)));

#define Bc  2
#define Sc  2048
#define Dc  1024
#define Hc  16
#define DKc 64

__device__ __forceinline__ v8f wmma_f32_k4(v2f a, v2f b, v8f c) {
  // D = A(16x4 f32) * B(4x16 f32) + C(16x16 f32)
  return __builtin_amdgcn_wmma_f32_16x16x4_f32(
      /*neg_a=*/false, a, /*neg_b=*/false, b,
      /*c_mod=*/(short)0, c, /*reuse_a=*/false, /*reuse_b=*/false);
}

// ---------------------------------------------------------------------------
// Kernel 1: Q/K/V = x @ W^T + b, RoPE on Q,K, stored as [B,H,S,DK]
// one wave computes one 16x16 output tile; blockIdx.z selects Q/K/V
// ---------------------------------------------------------------------------
__global__ __launch_bounds__(32) void qkv_rope_kernel(
    const float* __restrict__ x,
    const float* __restrict__ Wq, const float* __restrict__ bq,
    const float* __restrict__ Wk, const float* __restrict__ bk,
    const float* __restrict__ Wv, const float* __restrict__ bv,
    const int*   __restrict__ tokpos,
    float* __restrict__ qws, float* __restrict__ kws, float* __restrict__ vws)
{
  const int lane = threadIdx.x;
  const int m0   = blockIdx.x * 16;   // row tile over B*S
  const int n0   = blockIdx.y * 16;   // col tile over D
  const int kind = blockIdx.z;        // 0=Q 1=K 2=V (uniform per block)

  const float* W    = (kind == 0) ? Wq : (kind == 1) ? Wk : Wv;
  const float* bias = (kind == 0) ? bq : (kind == 1) ? bk : bv;
  float*       dst  = (kind == 0) ? qws : (kind == 1) ? kws : vws;

  const int mr = m0 + (lane & 15);
  const int n  = n0 + (lane & 15);
  const int kh = (lane >> 4) << 1;          // K sub-offset 0 or 2
  const float* arow = x + (size_t)mr * Dc;
  const float* brow = W + (size_t)n  * Dc;

  v8f acc = {};
  for (int k0 = 0; k0 < Dc; k0 += 4) {
    if ((k0 & 63) == 0) {
      __builtin_prefetch(arow + k0 + 64, 0, 1);
      __builtin_prefetch(brow + k0 + 64, 0, 1);
    }
    v2f a = *(const v2f*)(arow + k0 + kh);
    v2f b = *(const v2f*)(brow + k0 + kh);
    acc = wmma_f32_k4(a, b, acc);
  }

  const float bb    = bias[n];
  const int   mhalf = (lane >> 4) * 8;
  const int   h     = n >> 6;          // head
  const int   d     = n & (DKc - 1);   // dim within head

  #pragma unroll
  for (int v = 0; v < 8; ++v) {
    float val = acc[v] + bb;
    const int m = m0 + v + mhalf;
    const int s = m & (Sc - 1);
    if (kind < 2) {                    // RoPE on Q and K (uniform branch)
      const int   pos = tokpos[s];
      const int   dp  = d >> 1;        // frequency index
      const float inv = powf(10000.0f, -(2.0f * (float)dp) / (float)DKc);
      const float ang = (float)pos * inv;
      const float cv  = cosf(ang);
      const float sv  = sinf(ang);
      const float other = __shfl_xor(val, 1, 32);   // partner of the (even,odd) pair
      val = ((n & 1) == 0) ? (val * cv - other * sv)     // r1 = x1*c - x2*s
                           : (other * sv + val * cv);    // r2 = x1*s + x2*c
    }
    const int b = m >> 11;             // batch (S = 2048)
    dst[((size_t)((b * Hc + h) * Sc + s)) * DKc + d] = val;
  }
}

// ---------------------------------------------------------------------------
// Kernel 2: scores = Q @ K^T / sqrt(DK), causal mask -> attn buffer
// ---------------------------------------------------------------------------
__global__ __launch_bounds__(32) void scores_kernel(
    const float* __restrict__ qws, const float* __restrict__ kws,
    float* __restrict__ attn)
{
  const int lane = threadIdx.x;
  const int q0 = blockIdx.x * 16;
  const int k0 = blockIdx.y * 16;
  if (k0 > q0) return;                 // strictly-upper tile: softmax writes 0s
  const int bh = blockIdx.z;

  const float* Q = qws + (size_t)bh * Sc * DKc;
  const float* K = kws + (size_t)bh * Sc * DKc;
  const int kh = (lane >> 4) << 1;
  const float* qrow = Q + (size_t)(q0 + (lane & 15)) * DKc;
  const float* krow = K + (size_t)(k0 + (lane & 15)) * DKc;

  v8f acc = {};
  #pragma unroll
  for (int c = 0; c < DKc; c += 4) {
    v2f a = *(const v2f*)(qrow + c + kh);
    v2f b = *(const v2f*)(krow + c + kh);
    acc = wmma_f32_k4(a, b, acc);
  }

  const int n     = k0 + (lane & 15);
  const int mhalf = (lane >> 4) * 8;
  #pragma unroll
  for (int v = 0; v < 8; ++v) {
    const int m = q0 + v + mhalf;
    float sc = acc[v] * 0.125f;        // 1/sqrt(64)
    if (n > m) sc = -__builtin_inff();
    attn[((size_t)(bh * Sc + m)) * Sc + n] = sc;
  }
}

// ---------------------------------------------------------------------------
// Kernel 3: row softmax in place; zero-fill the causal tail
// ---------------------------------------------------------------------------
__global__ __launch_bounds__(256) void softmax_kernel(float* __restrict__ attn)
{
  const int row = blockIdx.x;          // 0 .. B*H*S-1
  const int q   = row & (Sc - 1);
  float* p = attn + (size_t)row * Sc;
  const int tid = threadIdx.x;
  __shared__ float red[8];

  float mx = -__builtin_inff();
  for (int k = tid; k <= q; k += 256) mx = fmaxf(mx, p[k]);
  for (int off = 16; off > 0; off >>= 1) mx = fmaxf(mx, __shfl_xor(mx, off, 32));
  if ((tid & 31) == 0) red[tid >> 5] = mx;
  __syncthreads();
  if (tid == 0) {
    float v = red[0];
    for (int i = 1; i < 8; ++i) v = fmaxf(v, red[i]);
    red[0] = v;
  }
  __syncthreads();
  mx = red[0];
  __syncthreads();

  float sum = 0.0f;
  for (int k = tid; k <= q; k += 256) {
    const float e = expf(p[k] - mx);
    p[k] = e;
    sum += e;
  }
  for (int off = 16; off > 0; off >>= 1) sum += __shfl_xor(sum, off, 32);
  if ((tid & 31) == 0) red[tid >> 5] = sum;
  __syncthreads();
  if (tid == 0) {
    float v = 0.0f;
    for (int i = 0; i < 8; ++i) v += red[i];
    red[0] = v;
  }
  __syncthreads();
  const float inv = 1.0f / red[0];

  for (int k = tid; k <= q; k += 256) p[k] *= inv;
  for (int k = q + 1 + tid; k < Sc; k += 256) p[k] = 0.0f;  // causal zeros
}

// ---------------------------------------------------------------------------
// Kernel 4: ctx = attn @ V, written in [B,S,D] layout (K-loop causally trimmed)
// ---------------------------------------------------------------------------
__global__ __launch_bounds__(32) void av_kernel(
    const float* __restrict__ attn, const float* __restrict__ vws,
    float* __restrict__ ctx)
{
  const int lane = threadIdx.x;
  const int q0 = blockIdx.x * 16;
  const int d0 = blockIdx.y * 16;
  const int bh = blockIdx.z;

  const float* P = attn + (size_t)(bh * Sc) * Sc;
  const float* V = vws  + (size_t)bh * Sc * DKc;
  const int kh = (lane >> 4) << 1;
  const float* prow = P + (size_t)(q0 + (lane & 15)) * Sc;
  const int d = d0 + (lane & 15);

  v8f acc = {};
  const int kend = q0 + 16;            // attn == 0 beyond the diagonal tile
  for (int c = 0; c < kend; c += 4) {
    v2f a = *(const v2f*)(prow + c + kh);
    v2f b = { V[(size_t)(c + kh)     * DKc + d],
              V[(size_t)(c + kh + 1) * DKc + d] };
    acc = wmma_f32_k4(a, b, acc);
  }

  const int bb = bh >> 4, h = bh & 15;
  const int mhalf = (lane >> 4) * 8;
  #pragma unroll
  for (int v = 0; v < 8; ++v) {
    const int qidx = q0 + v + mhalf;
    ctx[((size_t)(bb * Sc + qidx)) * Dc + h * DKc + d] = acc[v];
  }
}

// ---------------------------------------------------------------------------
// Kernel 5: out = ctx @ Wo^T + bo
// ---------------------------------------------------------------------------
__global__ __launch_bounds__(32) void outproj_kernel(
    const float* __restrict__ ctx, const float* __restrict__ Wo,
    const float* __restrict__ bo, float* __restrict__ out)
{
  const int lane = threadIdx.x;
  const int m0 = blockIdx.x * 16;
  const int n0 = blockIdx.y * 16;
  const int mr = m0 + (lane & 15);
  const int n  = n0 + (lane & 15);
  const int kh = (lane >> 4) << 1;
  const float* arow = ctx + (size_t)mr * Dc;
  const float* brow = Wo  + (size_t)n  * Dc;

  v8f acc = {};
  for (int k0 = 0; k0 < Dc; k0 += 4) {
    if ((k0 & 63) == 0) {
      __builtin_prefetch(arow + k0 + 64, 0, 1);
      __builtin_prefetch(brow + k0 + 64, 0, 1);
    }
    v2f a = *(const v2f*)(arow + k0 + kh);
    v2f b = *(const v2f*)(brow + k0 + kh);
    acc = wmma_f32_k4(a, b, acc);
  }

  const float bb = bo[n];
  const int mhalf = (lane >> 4) * 8;
  #pragma unroll
  for (int v = 0; v < 8; ++v)
    out[(size_t)(m0 + v + mhalf) * Dc + n] = acc[v] + bb;
}

// ---------------------------------------------------------------------------
extern "C" void kernel_launch(void* const* d_in, const int* in_sizes, int n_in,
                              void* d_out, int out_size, void* d_ws, size_t ws_size,
                              hipStream_t stream) {
  const float* x      = (const float*)d_in[0];
  const float* Wq     = (const float*)d_in[1];
  const float* bq     = (const float*)d_in[2];
  const float* Wk     = (const float*)d_in[3];
  const float* bk     = (const float*)d_in[4];
  const float* Wv     = (const float*)d_in[5];
  const float* bv     = (const float*)d_in[6];
  const float* Wo     = (const float*)d_in[7];
  const float* bo     = (const float*)d_in[8];
  const int*   tokpos = (const int*)d_in[9];
  // d_in[10] (mask) is the causal tril -> implemented analytically

  float* out  = (float*)d_out;
  float* attn = out + (size_t)Bc * Sc * Dc;   // tuple: [out | attn]

  const size_t qkvN = (size_t)Bc * Sc * Dc;   // 4,194,304 floats each
  float* qws = (float*)d_ws;
  float* kws = qws + qkvN;
  float* vws = kws + qkvN;                    // 48 MB total workspace
  float* ctx = qws;                           // Q dead after scores -> reuse

  dim3 g1(Bc * Sc / 16, Dc / 16, 3);
  qkv_rope_kernel<<<g1, 32, 0, stream>>>(x, Wq, bq, Wk, bk, Wv, bv, tokpos,
                                         qws, kws, vws);

  dim3 g2(Sc / 16, Sc / 16, Bc * Hc);
  scores_kernel<<<g2, 32, 0, stream>>>(qws, kws, attn);

  softmax_kernel<<<dim3(Bc * Hc * Sc), 256, 0, stream>>>(attn);

  dim3 g4(Sc / 16, DKc / 16, Bc * Hc);
  av_kernel<<<g4, 32, 0, stream>>>(attn, vws, ctx);

  dim3 g5(Bc * Sc / 16, Dc / 16);
  outproj_kernel<<<g5, 32, 0, stream>>>(ctx, Wo, bo, out);
}